// FeatureExtractor_50165218017745
// MI455X (gfx1250) — compile-verified
//
#include <hip/hip_runtime.h>

typedef float f4  __attribute__((ext_vector_type(4)));
typedef float v2f __attribute__((ext_vector_type(2)));
typedef float v8f __attribute__((ext_vector_type(8)));

#define BINS 32

// --- Kernel 1: per-(batch,quadrant) 32-bin histogram ------------------------
// grid = 256 blocks (b*4 + qy*2 + qx), block = 512 threads (16 waves, wave32).
// Per-wave x2 lane-parity LDS sub-histograms -> 32 sub-hists spanning all 64
// LDS banks; ds_add_u32 conflicts limited to <=16 lanes over 32 bins.

__device__ __forceinline__ void hist_add(float x, unsigned* __restrict__ h) {
  int bi = (int)(x * 32.0f);                 // trunc == floor for the valid range
  bi = bi < 0 ? 0 : (bi > 31 ? 31 : bi);
  if (x >= 0.0f && x <= 1.0f) atomicAdd(h + bi, 1u);
}

__global__ __launch_bounds__(512) void hist_regions_kernel(
    const float* __restrict__ in, float* __restrict__ ws_counts) {
  __shared__ unsigned hist[32 * BINS];       // 16 waves * 2 sub-hists
  const int tid = threadIdx.x;
  hist[tid] = 0u;
  hist[tid + 512] = 0u;
  __syncthreads();

  const int region = blockIdx.x;             // b*4 + qy*2 + qx
  const int b  = region >> 2;
  const int qy = (region >> 1) & 1;
  const int qx = region & 1;
  const float* base = in + (size_t)b * (3u * 512u * 512u);

  unsigned* h = hist + (((tid >> 5) * 2 + (tid & 1)) * BINS);

  // Region = 3 channels x 256 rows x 256 cols = 49152 float4 (49152 % 2048 == 0)
  for (int i = tid; i < 49152; i += 2048) {
    f4 v[4];
#pragma unroll
    for (int u = 0; u < 4; ++u) {            // 4 B128 loads in flight / thread
      int idx = i + u * 512;
      int ww4 = idx & 63;                    // float4 within 256-wide row half
      int row = idx >> 6;                    // 0..767 == c*256 + hh
      int c   = row >> 8;
      int hh  = row & 255;
      size_t off = ((size_t)(c * 512 + qy * 256 + hh) * 512)
                 + (size_t)(qx * 256 + ww4 * 4);
      v[u] = __builtin_nontemporal_load((const f4*)(base + off));  // stream, don't thrash L2
    }
#pragma unroll
    for (int u = 0; u < 4; ++u) {
      hist_add(v[u].x, h);
      hist_add(v[u].y, h);
      hist_add(v[u].z, h);
      hist_add(v[u].w, h);
    }
  }
  __syncthreads();

  if (tid < BINS) {                          // reduce 32 sub-histograms
    unsigned s = 0;
#pragma unroll
    for (int k = 0; k < 32; ++k) s += hist[k * BINS + tid];
    ws_counts[region * BINS + tid] = (float)s;
  }
}

// --- Kernel 2: finalize -----------------------------------------------------
// grid = 64 blocks (one per batch), block = 128 threads.
// s0[bin] = sum_q counts[q][bin] done as a real 16x4 x 4x16 WMMA against a
// ones matrix (V_WMMA_F32_16X16X4_F32); the D fragment's 16 replicated lane
// copies ARE the 4x4 spatial broadcast, stored straight out.

__global__ __launch_bounds__(128) void finalize_kernel(
    const float* __restrict__ ws_counts, float* __restrict__ out) {
  const int b   = blockIdx.x;
  const int tid = threadIdx.x;
  const float* cw = ws_counts + b * 4 * BINS;   // [q][32], q = qy*2+qx
  float* ob = out + (size_t)b * 1536;           // 96 bins * 16 cells

  if (tid < 32) {                               // wave 0, EXEC all-1s
    const int lane = tid;
    const int qb   = (lane >> 4) * 2;           // A 16x4 f32 layout: lanes 0-15
    const int binl = lane & 15;                 // hold K=0,1; lanes 16-31 K=2,3
    const float inv_tot = 1.0f / 262144.0f;     // 1/(512*512)
    v2f ones = {1.0f, 1.0f};

    v2f a0, a1;
    a0.x = cw[qb * 32 + binl]            * inv_tot;   // A[m=bin][k=qb]
    a0.y = cw[(qb + 1) * 32 + binl]      * inv_tot;   // A[m=bin][k=qb+1]
    a1.x = cw[qb * 32 + 16 + binl]       * inv_tot;   // bins 16..31
    a1.y = cw[(qb + 1) * 32 + 16 + binl] * inv_tot;

    v8f c0 = {};
    v8f c1 = {};
    c0 = __builtin_amdgcn_wmma_f32_16x16x4_f32(false, a0, false, ones,
                                               (short)0, c0, false, false);
    c1 = __builtin_amdgcn_wmma_f32_16x16x4_f32(false, a1, false, ones,
                                               (short)0, c1, false, false);

    // D layout: lanes 0-15 vgpr r -> M=r (bin r), lanes 16-31 -> M=r+8; N=lane&15.
    const int mhi  = (lane >> 4) * 8;
    const int cell = lane & 15;
#pragma unroll
    for (int r = 0; r < 8; ++r) {
      ob[(r + mhi) * 16 + cell]      = c0[r];   // h0, bins 0..15
      ob[(16 + r + mhi) * 16 + cell] = c1[r];   // h0, bins 16..31
    }
  }

  // h1: quadrant hist / 65536 repeated over its 2x2 block; h2: zeros.
  const float inv_q = 1.0f / 65536.0f;          // 1/(256*256)
  for (int idx = tid; idx < 512; idx += 128) {
    int bin  = idx >> 4;
    int cell = idx & 15;
    int y = cell >> 2, x = cell & 3;
    int q = (y >> 1) * 2 + (x >> 1);
    ob[512 + idx]  = cw[q * 32 + bin] * inv_q;  // h1
    ob[1024 + idx] = 0.0f;                      // h2
  }
}

extern "C" void kernel_launch(void* const* d_in, const int* in_sizes, int n_in,
                              void* d_out, int out_size, void* d_ws, size_t ws_size,
                              hipStream_t stream) {
  const float* in  = (const float*)d_in[0];     // [64,3,512,512] fp32
  float* out = (float*)d_out;                   // [64,96,4,4] fp32
  float* ws  = (float*)d_ws;                    // 256*32 floats = 32 KB counts

  hist_regions_kernel<<<256, 512, 0, stream>>>(in, ws);
  finalize_kernel<<<64, 128, 0, stream>>>(ws, out);
}